// LSHOrdering_45732811768046
// MI455X (gfx1250) — compile-verified
//
#include <hip/hip_runtime.h>

#define B_ 8
#define L_ 32768
#define D_ 128

typedef __attribute__((ext_vector_type(2))) float v2f;
typedef __attribute__((ext_vector_type(8))) float v8f;

// ---------------------------------------------------------------------------
// Kernel 1: per-token bucket key (via WMMA f32 16x16x4) + norm -> sortable
// 64-bit key and index payload.
// One wave (32 lanes) handles 16 tokens. Block = 256 threads = 8 waves = 128
// tokens. Grid = B*L/128 blocks.
// ---------------------------------------------------------------------------
__global__ void lsh_keys_kernel(const float* __restrict__ points,
                                const float* __restrict__ alpha,
                                unsigned long long* __restrict__ keys,
                                unsigned int* __restrict__ idxs) {
    __shared__ float sh_w[D_];
    __shared__ float sh_bk[8 * 16];

    const int tid = threadIdx.x;

    // w[d] = alpha[d,0] + 2*alpha[d,1] + 4*alpha[d,2]   (alpha row-major D x 3)
    if (tid < D_) {
        float a0 = alpha[tid * 3 + 0];
        float a1 = alpha[tid * 3 + 1];
        float a2 = alpha[tid * 3 + 2];
        sh_w[tid] = a0 + 2.0f * a1 + 4.0f * a2;
    }
    __syncthreads();

    const int wave   = tid >> 5;
    const int lane   = tid & 31;
    const int laneLo = lane & 15;
    const bool hi    = lane >= 16;

    const long long t0  = (long long)blockIdx.x * 128 + (long long)wave * 16;
    const long long row = t0 + laneLo;           // this lane's token (M = laneLo)
    const float* prow   = points + row * D_;

    v8f  c = {};          // 16x16 f32 accumulator (all N columns identical)
    float sumsq = 0.0f;

    // A layout (32-bit A 16x4): lanes 0-15 hold K=k0,k0+1 ; lanes 16-31 K=k0+2,k0+3
    // B layout (4x16, all columns = w chunk): mirrors the same per-lane K split.
    for (int k0 = 0; k0 < D_; k0 += 4) {
        const int col = k0 + (hi ? 2 : 0);
        v2f a;
        a.x = prow[col];
        a.y = prow[col + 1];
        v2f b;
        b.x = sh_w[col];
        b.y = sh_w[col + 1];
        sumsq += a.x * a.x + a.y * a.y;
        // D = A(16x4) * B(4x16) + C ; accumulate over K chunks
        c = __builtin_amdgcn_wmma_f32_16x16x4_f32(
                false, a, false, b, (short)0, c, false, false);
    }

    // norm^2: lane l (K even halves) + lane l^16 (K odd halves)
    sumsq += __shfl_xor(sumsq, 16, 32);

    // C/D layout: VGPR r -> M=r (lanes 0-15), M=8+r (lanes 16-31). Every column
    // holds the same value (B columns identical), so lane 0 / lane 16 publish
    // the 16 per-token bucket keys.
    if (lane == 0 || lane == 16) {
        float* dst = &sh_bk[wave * 16 + (hi ? 8 : 0)];
        dst[0] = c[0]; dst[1] = c[1]; dst[2] = c[2]; dst[3] = c[3];
        dst[4] = c[4]; dst[5] = c[5]; dst[6] = c[6]; dst[7] = c[7];
    }
    __syncthreads();

    if (!hi) {
        const float bk  = sh_bk[wave * 16 + laneLo];
        const float nrm = sqrtf(sumsq);

        unsigned ub = __float_as_uint(bk);
        ub = (ub & 0x80000000u) ? ~ub : (ub | 0x80000000u);   // total-order float
        unsigned un = __float_as_uint(nrm) | 0x80000000u;     // nrm >= 0

        keys[row] = ((unsigned long long)ub << 32) | (unsigned long long)un;
        idxs[row] = (unsigned)(row & (L_ - 1));               // index within batch
    }
}

// ---------------------------------------------------------------------------
// Bitonic sort helpers. Comparator: (key, idx) lexicographic -> reproduces the
// reference's chained stable sorts exactly. Directions use within-batch index
// so the 8 batches sort independently (all ascending).
// ---------------------------------------------------------------------------
__device__ __forceinline__ bool kgt(unsigned long long ka, unsigned ia,
                                    unsigned long long kb, unsigned ib) {
    return (ka > kb) || (ka == kb && ia > ib);
}

// Fully sorts each 2048-element chunk in LDS (stages k = 2 .. 2048).
__global__ void bitonic_local_sort(unsigned long long* __restrict__ keys,
                                   unsigned int* __restrict__ idxs) {
    __shared__ unsigned long long sk[2048];
    __shared__ unsigned int       si[2048];
    const int t = threadIdx.x;                     // 0..1023
    const long long base = (long long)blockIdx.x * 2048;

    sk[t]        = keys[base + t];        si[t]        = idxs[base + t];
    sk[t + 1024] = keys[base + t + 1024]; si[t + 1024] = idxs[base + t + 1024];
    __syncthreads();

    for (int k = 2; k <= 2048; k <<= 1) {
        for (int j = k >> 1; j >= 1; j >>= 1) {
            const int i = (t << 1) - (t & (j - 1));  // bit log2(j) of i is 0
            const int p = i, q = i + j;
            const int gib = (int)((base + i) & (L_ - 1));
            const bool up = ((gib & k) == 0);
            unsigned long long ka = sk[p], kb = sk[q];
            unsigned ia = si[p], ib = si[q];
            if (kgt(ka, ia, kb, ib) == up) {
                sk[p] = kb; sk[q] = ka; si[p] = ib; si[q] = ia;
            }
            __syncthreads();
        }
    }
    keys[base + t]        = sk[t];        idxs[base + t]        = si[t];
    keys[base + t + 1024] = sk[t + 1024]; idxs[base + t + 1024] = si[t + 1024];
}

// One global compare-exchange pass for stage (k, j), j >= 2048.
__global__ void bitonic_global_pass(unsigned long long* __restrict__ keys,
                                    unsigned int* __restrict__ idxs,
                                    int j, int k) {
    const long long t = (long long)blockIdx.x * blockDim.x + threadIdx.x; // B*L/2
    const long long i = (t << 1) - (t & (long long)(j - 1));
    const long long q = i + j;
    const int gib = (int)(i & (L_ - 1));
    const bool up = ((gib & k) == 0);
    unsigned long long ka = keys[i], kb = keys[q];
    unsigned ia = idxs[i], ib = idxs[q];
    if (kgt(ka, ia, kb, ib) == up) {
        keys[i] = kb; keys[q] = ka; idxs[i] = ib; idxs[q] = ia;
    }
}

// Finishes stage k: runs j = 1024 .. 1 inside LDS on each 2048-element chunk.
__global__ void bitonic_local_merge(unsigned long long* __restrict__ keys,
                                    unsigned int* __restrict__ idxs, int k) {
    __shared__ unsigned long long sk[2048];
    __shared__ unsigned int       si[2048];
    const int t = threadIdx.x;
    const long long base = (long long)blockIdx.x * 2048;

    sk[t]        = keys[base + t];        si[t]        = idxs[base + t];
    sk[t + 1024] = keys[base + t + 1024]; si[t + 1024] = idxs[base + t + 1024];
    __syncthreads();

    const bool up = ((((int)(base & (L_ - 1))) & k) == 0); // constant per chunk
    for (int j = 1024; j >= 1; j >>= 1) {
        const int i = (t << 1) - (t & (j - 1));
        const int p = i, q = i + j;
        unsigned long long ka = sk[p], kb = sk[q];
        unsigned ia = si[p], ib = si[q];
        if (kgt(ka, ia, kb, ib) == up) {
            sk[p] = kb; sk[q] = ka; si[p] = ib; si[q] = ia;
        }
        __syncthreads();
    }
    keys[base + t]        = sk[t];        idxs[base + t]        = si[t];
    keys[base + t + 1024] = sk[t + 1024]; idxs[base + t + 1024] = si[t + 1024];
}

// ---------------------------------------------------------------------------
// Gather: ordered_points[b,i,:] = points[b, idx, :] (float4 per thread) and
// emit int32 indices after the B*L*D float region of d_out.
// ---------------------------------------------------------------------------
__global__ void gather_kernel(const float* __restrict__ points,
                              const unsigned int* __restrict__ idxs,
                              float* __restrict__ outp,
                              int* __restrict__ outi) {
    const long long g   = (long long)blockIdx.x * blockDim.x + threadIdx.x;
    const long long tok = g >> 5;                 // output token (global)
    const int       q   = (int)(g & 31);          // which float4 of D=128
    const long long base = tok & ~(long long)(L_ - 1);   // batch*L
    const unsigned src  = idxs[tok];

    const float4* s = (const float4*)(points + (base + (long long)src) * D_) + q;
    float4*       d = (float4*)(outp + tok * D_) + q;
    *d = *s;
    if (q == 0) outi[tok] = (int)src;
}

// ---------------------------------------------------------------------------
extern "C" void kernel_launch(void* const* d_in, const int* in_sizes, int n_in,
                              void* d_out, int out_size, void* d_ws, size_t ws_size,
                              hipStream_t stream) {
    (void)in_sizes; (void)n_in; (void)out_size; (void)ws_size;

    const float* points = (const float*)d_in[0];
    const float* alpha  = (const float*)d_in[1];

    float* outp = (float*)d_out;
    int*   outi = (int*)((float*)d_out + (size_t)B_ * L_ * D_);

    unsigned long long* keys = (unsigned long long*)d_ws;
    unsigned int*       idxs =
        (unsigned int*)((char*)d_ws + (size_t)B_ * L_ * sizeof(unsigned long long));

    // 1) keys (WMMA projection + norms)
    lsh_keys_kernel<<<(B_ * L_) / 128, 256, 0, stream>>>(points, alpha, keys, idxs);

    // 2) bitonic sort: local full sort of 2048-chunks, then global merges with
    //    LDS-fused tails (j <= 1024).
    bitonic_local_sort<<<(B_ * L_) / 2048, 1024, 0, stream>>>(keys, idxs);
    for (int k = 4096; k <= L_; k <<= 1) {
        for (int j = k >> 1; j >= 2048; j >>= 1) {
            bitonic_global_pass<<<(B_ * L_ / 2) / 256, 256, 0, stream>>>(keys, idxs, j, k);
        }
        bitonic_local_merge<<<(B_ * L_) / 2048, 1024, 0, stream>>>(keys, idxs, k);
    }

    // 3) gather output points + indices
    gather_kernel<<<(B_ * L_ * (D_ / 4)) / 256, 256, 0, stream>>>(points, idxs, outp, outi);
}